// Model_28681791602783
// MI455X (gfx1250) — compile-verified
//
#include <hip/hip_runtime.h>
#include <hip/hip_bf16.h>
#include <stdint.h>

#ifndef __has_builtin
#define __has_builtin(x) 0
#endif

// Vector type matching the builtin signature leaked by the round-1 diagnostic:
// 'int __attribute__((__vector_size__(4 * sizeof(int)))) __device__ *'
typedef int v4i __attribute__((vector_size(16)));

// ---------------------------------------------------------------------------
// gfx1250 async Global<->LDS copy support (ASYNCcnt-tracked DMA-style path).
// ---------------------------------------------------------------------------
#if defined(__gfx1250__) && \
    __has_builtin(__builtin_amdgcn_global_load_async_to_lds_b128) && \
    __has_builtin(__builtin_amdgcn_global_store_async_from_lds_b128)
#define USE_ASYNC_LDS_COPY 1
#else
#define USE_ASYNC_LDS_COPY 0
#endif

#if __has_builtin(__builtin_amdgcn_s_wait_asynccnt)
#define WAIT_ASYNCCNT_0() __builtin_amdgcn_s_wait_asynccnt(0)
#else
#define WAIT_ASYNCCNT_0() asm volatile("s_wait_asynccnt 0" ::: "memory")
#endif

#define COPY_BLOCK 256   // 8 wave32s per workgroup
#define UNROLL 4         // 4 async B128 ops in flight per lane per phase

// ---------------------------------------------------------------------------
// Kernel 1: bulk copy of block_tables (int32, 8192*4096 elems) -> d_out.
// Pure HBM-bandwidth bound: 16B/lane per async op, batched x4, grid-stride.
// ---------------------------------------------------------------------------
__global__ __launch_bounds__(COPY_BLOCK) void bt_copy_kernel(
    const v4i* __restrict__ src, v4i* __restrict__ dst, long long n16)
{
#if USE_ASYNC_LDS_COPY
    // 16 KB LDS staging tile; each lane owns UNROLL private 16B slots.
    // Async engine moves global -> LDS -> global, no VGPR data movement.
    __shared__ v4i tile[UNROLL * COPY_BLOCK];
    const int tid = threadIdx.x;
    const long long chunk  = (long long)UNROLL * COPY_BLOCK;
    const long long stride = (long long)gridDim.x * chunk;
    for (long long base = (long long)blockIdx.x * chunk; base < n16;
         base += stride) {
#pragma unroll
        for (int u = 0; u < UNROLL; ++u) {
            const long long i = base + (long long)u * COPY_BLOCK + tid;
            if (i < n16) {
                __builtin_amdgcn_global_load_async_to_lds_b128(
                    (v4i*)(src + i), &tile[u * COPY_BLOCK + tid], 0, 0);
            }
        }
        WAIT_ASYNCCNT_0();                 // all 4 loads landed in LDS
#pragma unroll
        for (int u = 0; u < UNROLL; ++u) {
            const long long i = base + (long long)u * COPY_BLOCK + tid;
            if (i < n16) {
                __builtin_amdgcn_global_store_async_from_lds_b128(
                    (v4i*)(dst + i), &tile[u * COPY_BLOCK + tid], 0, 0);
            }
        }
        WAIT_ASYNCCNT_0();                 // LDS slots reusable next iter
    }
#else
    long long i = (long long)blockIdx.x * COPY_BLOCK + threadIdx.x;
    const long long stride = (long long)gridDim.x * COPY_BLOCK;
    for (; i < n16; i += stride) {
        dst[i] = src[i];                   // global_load_b128 / global_store_b128
    }
#endif
}

// ---------------------------------------------------------------------------
// Kernel 2: copy num_blocks[0, :] (MAX_REQS ints) into output tail.
// ---------------------------------------------------------------------------
__global__ void nb_copy_kernel(const int* __restrict__ src,
                               int* __restrict__ dst, int n)
{
    int i = blockIdx.x * blockDim.x + threadIdx.x;
    if (i < n) dst[i] = src[i];
}

// ---------------------------------------------------------------------------
// Kernel 3: scatter the appended block ids. One 64-thread block (2 waves)
// per request; destination columns are contiguous -> fully coalesced.
// Runs after kernels 1 & 2 (same stream => ordered).
// ---------------------------------------------------------------------------
__global__ __launch_bounds__(64) void scatter_kernel(
    const int* __restrict__ req_indices,         // [num_reqs]
    const int* __restrict__ cu_num_new,          // [num_reqs+1] (group 0)
    const int* __restrict__ new_block_ids,       // [total]      (group 0)
    const unsigned char* __restrict__ overwrite, // [num_reqs] (bool)
    const int* __restrict__ num_blocks_in,       // [MAX_REQS]   (group 0)
    int* __restrict__ out_bt,                    // [MAX_REQS * stride]
    int* __restrict__ out_nb,                    // [MAX_REQS]
    int row_stride)
{
    const int b   = blockIdx.x;
    const int cu0 = cu_num_new[b];
    const int cu1 = cu_num_new[b + 1];
    const int cnt = cu1 - cu0;
    const int req = req_indices[b];
    const int dst_start = overwrite[b] ? 0 : num_blocks_in[req];

    int* row = out_bt + (long long)req * row_stride + dst_start;
    const int* srcp = new_block_ids + cu0;
    for (int k = threadIdx.x; k < cnt; k += 64) {
        row[k] = srcp[k];
    }
    if (threadIdx.x == 0) {
        out_nb[req] = dst_start + cnt;
    }
}

// ---------------------------------------------------------------------------
// Launch
// ---------------------------------------------------------------------------
extern "C" void kernel_launch(void* const* d_in, const int* in_sizes, int n_in,
                              void* d_out, int out_size, void* d_ws, size_t ws_size,
                              hipStream_t stream)
{
    (void)n_in; (void)d_ws; (void)ws_size; (void)out_size;

    const int*           req_indices   = (const int*)d_in[0];
    const int*           cu_num_new    = (const int*)d_in[1];   // [1, num_reqs+1]
    const int*           new_block_ids = (const int*)d_in[2];   // [1, total]
    const unsigned char* overwrite     = (const unsigned char*)d_in[3];
    const int*           num_blocks    = (const int*)d_in[6];   // [1, MAX_REQS]
    const int*           block_tables  = (const int*)d_in[7];   // [MAX_REQS * MAX_BLOCKS]

    const int num_reqs   = in_sizes[0];
    const int nb_elems   = in_sizes[6];                 // MAX_REQS (G==1)
    const long long bt_elems = (long long)in_sizes[7];
    const int row_stride = (int)(bt_elems / nb_elems);  // MAX_BLOCKS

    int* out_bt = (int*)d_out;
    int* out_nb = out_bt + bt_elems;

    // 1) 128 MB pass-through copy (HBM-bound: ~11 us at 23.3 TB/s).
    const long long n16 = bt_elems / 4;                 // 16B elements
    const long long chunk = (long long)UNROLL * COPY_BLOCK;
    long long want = (n16 + chunk - 1) / chunk;
    int grid = (int)(want < 8192 ? want : 8192);        // grid-stride beyond this
    bt_copy_kernel<<<grid, COPY_BLOCK, 0, stream>>>(
        (const v4i*)block_tables, (v4i*)out_bt, n16);

    // 2) num_blocks copy.
    nb_copy_kernel<<<(nb_elems + 255) / 256, 256, 0, stream>>>(
        num_blocks, out_nb, nb_elems);

    // 3) scattered appends (after copies; same stream => ordered).
    scatter_kernel<<<num_reqs, 64, 0, stream>>>(
        req_indices, cu_num_new, new_block_ids, overwrite,
        num_blocks, out_bt, out_nb, row_stride);
}